// DOMINANTAugmented_61512521613989
// MI455X (gfx1250) — compile-verified
//
#include <hip/hip_runtime.h>

// ---------------------------------------------------------------------------
// DOMINANT graph autoencoder, CDNA5 (gfx1250) implementation.
//   N_NODES=12288, N_EDGES=393216, IN_DIM=500, HID_DIM=64
// Outputs (concatenated in d_out): x_hat [12288*500], s_hat [12288*12288]
// ---------------------------------------------------------------------------

#define N_NODES 12288
#define N_EDGES 393216
#define IN_DIM  500
#define HID_DIM 64

typedef __attribute__((ext_vector_type(2))) float v2f;
typedef __attribute__((ext_vector_type(8))) float v8f;

// ---------------------------------------------------------------------------
// fp32 WMMA GEMM:  C[M x N] = A[M x K] @ B   (B row-major [K x N]; if TRANSB,
// B is [N x K] row-major and we compute A @ B^T).
// One wave32 computes one 16x16 C tile with V_WMMA_F32_16X16X4_F32.
// Block = 256 threads = 8 waves -> 32x64 C tile per block.
//
// No bounds branches in the K loop: out-of-range lanes clamp their row/col
// index to the last valid one and load real data unconditionally; the
// resulting garbage only affects C elements that are never stored (stores are
// predicated after all WMMAs complete). K must be a multiple of 4.
// KFIX != 0 makes K a compile-time constant (full unroll, back-to-back WMMA).
// ---------------------------------------------------------------------------
template <int TRANSB, int NTSTORE, int KFIX>
__global__ __launch_bounds__(256)
void gemm_wmma_f32(const float* __restrict__ A, const float* __restrict__ B,
                   float* __restrict__ C, int M, int N, int Krt,
                   int lda, int ldb, int ldc)
{
    const int K    = KFIX ? KFIX : Krt;
    const int lane = threadIdx.x & 31;
    const int wave = threadIdx.x >> 5;     // 0..7
    const int l16  = lane & 15;
    const int half = lane >> 4;            // 0: K{0,1}, 1: K{2,3} fragment half

    const int rowBase = blockIdx.y * 32 + (wave >> 2) * 16;
    const int colBase = blockIdx.x * 64 + (wave & 3) * 16;

    const int aRowS = rowBase + l16;                 // true row (for store pred)
    const int bColS = colBase + l16;                 // true col (for store pred)
    const int aRow  = aRowS < M ? aRowS : (M - 1);   // clamped: load-safe
    const int bCol  = bColS < N ? bColS : (N - 1);

    const float* Ap = A + (size_t)aRow * lda + half * 2;     // contiguous pair
    const float* Bp = TRANSB ? (B + (size_t)bCol * ldb + half * 2)
                             : (B + (size_t)(half * 2) * ldb + bCol);

    v8f acc = {};
    if (KFIX) {
#pragma unroll
        for (int k0 = 0; k0 < KFIX; k0 += 4) {
            v2f a = *(const v2f*)(Ap + k0);
            v2f b;
            if (TRANSB) {
                b = *(const v2f*)(Bp + k0);
            } else {
                b.x = Bp[(size_t)k0 * ldb];
                b.y = Bp[(size_t)(k0 + 1) * ldb];
            }
            acc = __builtin_amdgcn_wmma_f32_16x16x4_f32(
                      false, a, false, b, (short)0, acc, false, false);
        }
    } else {
#pragma unroll 4
        for (int k0 = 0; k0 < K; k0 += 4) {
            __builtin_prefetch(Ap + k0 + 64, 0, 1);  // global_prefetch_b8
            v2f a = *(const v2f*)(Ap + k0);
            v2f b;
            if (TRANSB) {
                b = *(const v2f*)(Bp + k0);
            } else {
                b.x = Bp[(size_t)k0 * ldb];
                b.y = Bp[(size_t)(k0 + 1) * ldb];
            }
            acc = __builtin_amdgcn_wmma_f32_16x16x4_f32(
                      false, a, false, b, (short)0, acc, false, false);
        }
    }

    // C/D layout: VGPR r -> row (half*8 + r), col l16
    if (bColS < N) {
#pragma unroll
        for (int r = 0; r < 8; ++r) {
            const int row = rowBase + half * 8 + r;
            if (row < M) {
                float* p = &C[(size_t)row * ldc + bColS];
                if (NTSTORE) __builtin_nontemporal_store(acc[r], p);
                else         *p = acc[r];
            }
        }
    }
}

// ---------------------------------------------------------------------------
// Helper kernels
// ---------------------------------------------------------------------------
__global__ void fill_f32(float* __restrict__ p, float v, int n) {
    int i = blockIdx.x * blockDim.x + threadIdx.x;
    if (i < n) p[i] = v;
}

// in-degree accumulation over dst[] (self-loop "1" pre-seeded by fill)
__global__ void deg_accum(const int* __restrict__ dst, float* __restrict__ deg, int e) {
    int i = blockIdx.x * blockDim.x + threadIdx.x;
    if (i < e) atomicAdd(&deg[dst[i]], 1.0f);
}

__global__ void rsqrt_inplace(float* __restrict__ p, int n) {
    int i = blockIdx.x * blockDim.x + threadIdx.x;
    if (i < n) p[i] = __frsqrt_rn(p[i]);
}

// Scatter-add message passing: out[dst] += h[src] * dis[src]*dis[dst]
// One wave32 per (virtual) edge; lanes cover features as float2 chunks.
// Virtual edge list: e < E -> graph edge; e >= E -> self-loop (src=dst=e-E).
__global__ __launch_bounds__(256)
void propagate(const float* __restrict__ h, float* __restrict__ out,
               const int* __restrict__ src, const int* __restrict__ dst,
               const float* __restrict__ dis, int E, int n, int dim)
{
    const int wid  = (int)((blockIdx.x * (long long)blockDim.x + threadIdx.x) >> 5);
    const int lane = threadIdx.x & 31;
    if (wid >= E + n) return;

    int s, d;
    if (wid < E) { s = src[wid]; d = dst[wid]; }
    else         { s = d = wid - E; }

    const float   nrm = dis[s] * dis[d];
    const float2* hp  = (const float2*)(h + (size_t)s * dim);
    float*        op  = out + (size_t)d * dim;

    const int dim2 = dim >> 1;
    for (int c = lane; c < dim2; c += 32) {
        float2 v = hp[c];
        atomicAdd(op + 2 * c,     v.x * nrm);
        atomicAdd(op + 2 * c + 1, v.y * nrm);
    }
}

// out[i,j] += b[j]; optional ReLU
__global__ void bias_act(float* __restrict__ out, const float* __restrict__ b,
                         int n, int dim, int relu)
{
    long long i = (long long)blockIdx.x * blockDim.x + threadIdx.x;
    long long tot = (long long)n * dim;
    if (i >= tot) return;
    float v = out[i] + b[i % dim];
    if (relu) v = fmaxf(v, 0.0f);
    out[i] = v;
}

// ---------------------------------------------------------------------------
// Host orchestration
// ---------------------------------------------------------------------------
static inline int cdiv(long long a, int b) { return (int)((a + b - 1) / b); }

extern "C" void kernel_launch(void* const* d_in, const int* in_sizes, int n_in,
                              void* d_out, int out_size, void* d_ws, size_t ws_size,
                              hipStream_t stream)
{
    const float* x       = (const float*)d_in[0];
    const int*   src     = (const int*)  d_in[1];
    const int*   dst     = (const int*)  d_in[2];
    const float* enc_w1  = (const float*)d_in[3];
    const float* enc_b1  = (const float*)d_in[4];
    const float* enc_w2  = (const float*)d_in[5];
    const float* enc_b2  = (const float*)d_in[6];
    const float* attr_w1 = (const float*)d_in[7];
    const float* attr_b1 = (const float*)d_in[8];
    const float* attr_w2 = (const float*)d_in[9];
    const float* attr_b2 = (const float*)d_in[10];
    const float* str_w1  = (const float*)d_in[11];
    const float* str_b1  = (const float*)d_in[12];

    const int NN = N_NODES, E = N_EDGES, IN = IN_DIM, H = HID_DIM;

    float* x_hat = (float*)d_out;                         // [NN, IN]
    float* s_hat = x_hat + (size_t)NN * IN;               // [NN, NN]

    // workspace carve-up
    float* dis = (float*)d_ws;                            // [NN]
    float* tA  = dis + NN;                                // [NN, H] gemm out
    float* hA  = tA  + (size_t)NN * H;                    // [NN, H] propagated
    float* emb = hA  + (size_t)NN * H;                    // [NN, H]
    float* tB  = emb + (size_t)NN * H;                    // [NN, H]
    float* h2  = tB  + (size_t)NN * H;                    // [NN, H]
    float* z   = h2  + (size_t)NN * H;                    // [NN, H]
    float* big = z   + (size_t)NN * H;                    // [NN, IN]

    const int TB = 256;
    const dim3 blk(TB);

    // ---- degree -> dis = rsqrt(deg), deg seeded with 1.0 (self loop) ----
    fill_f32<<<cdiv(NN, TB), blk, 0, stream>>>(dis, 1.0f, NN);
    deg_accum<<<cdiv(E, TB), blk, 0, stream>>>(dst, dis, E);
    rsqrt_inplace<<<cdiv(NN, TB), blk, 0, stream>>>(dis, NN);

    // GEMM grid helper: 32 rows x 64 cols of C per block
    auto gemm_grid = [](int M, int N) { return dim3((N + 63) / 64, (M + 31) / 32); };
    const int prop_blocks = cdiv((long long)(E + NN) * 32, TB);  // wave per edge

    // ---- encoder layer 1: relu(prop(x @ enc_w1) + b1) ----
    gemm_wmma_f32<0,0,0><<<gemm_grid(NN, H), blk, 0, stream>>>(x, enc_w1, tA, NN, H, IN, IN, H, H);
    fill_f32<<<cdiv((long long)NN*H, TB), blk, 0, stream>>>(hA, 0.0f, NN*H);
    propagate<<<prop_blocks, blk, 0, stream>>>(tA, hA, src, dst, dis, E, NN, H);
    bias_act<<<cdiv((long long)NN*H, TB), blk, 0, stream>>>(hA, enc_b1, NN, H, 1);

    // ---- encoder layer 2: emb = prop(hA @ enc_w2) + b2 ----
    gemm_wmma_f32<0,0,HID_DIM><<<gemm_grid(NN, H), blk, 0, stream>>>(hA, enc_w2, tB, NN, H, H, H, H, H);
    fill_f32<<<cdiv((long long)NN*H, TB), blk, 0, stream>>>(emb, 0.0f, NN*H);
    propagate<<<prop_blocks, blk, 0, stream>>>(tB, emb, src, dst, dis, E, NN, H);
    bias_act<<<cdiv((long long)NN*H, TB), blk, 0, stream>>>(emb, enc_b2, NN, H, 0);

    // ---- attribute decoder layer 1: h2 = relu(prop(emb @ attr_w1) + b) ----
    gemm_wmma_f32<0,0,HID_DIM><<<gemm_grid(NN, H), blk, 0, stream>>>(emb, attr_w1, tA, NN, H, H, H, H, H);
    fill_f32<<<cdiv((long long)NN*H, TB), blk, 0, stream>>>(h2, 0.0f, NN*H);
    propagate<<<prop_blocks, blk, 0, stream>>>(tA, h2, src, dst, dis, E, NN, H);
    bias_act<<<cdiv((long long)NN*H, TB), blk, 0, stream>>>(h2, attr_b1, NN, H, 1);

    // ---- attribute decoder layer 2: x_hat = prop(h2 @ attr_w2) + b ----
    gemm_wmma_f32<0,0,HID_DIM><<<gemm_grid(NN, IN), blk, 0, stream>>>(h2, attr_w2, big, NN, IN, H, H, IN, IN);
    fill_f32<<<cdiv((long long)NN*IN, TB), blk, 0, stream>>>(x_hat, 0.0f, NN*IN);
    propagate<<<prop_blocks, blk, 0, stream>>>(big, x_hat, src, dst, dis, E, NN, IN);
    bias_act<<<cdiv((long long)NN*IN, TB), blk, 0, stream>>>(x_hat, attr_b2, NN, IN, 0);

    // ---- structure decoder: z = prop(emb @ str_w1) + b ----
    gemm_wmma_f32<0,0,HID_DIM><<<gemm_grid(NN, H), blk, 0, stream>>>(emb, str_w1, tB, NN, H, H, H, H, H);
    fill_f32<<<cdiv((long long)NN*H, TB), blk, 0, stream>>>(z, 0.0f, NN*H);
    propagate<<<prop_blocks, blk, 0, stream>>>(tB, z, src, dst, dis, E, NN, H);
    bias_act<<<cdiv((long long)NN*H, TB), blk, 0, stream>>>(z, str_b1, NN, H, 0);

    // ---- s_hat = z @ z^T  (store-bandwidth bound: 604 MB -> NT stores) ----
    gemm_wmma_f32<1,1,HID_DIM><<<gemm_grid(NN, NN), blk, 0, stream>>>(z, z, s_hat, NN, NN, H, H, H, NN);
}